// Edge_Relation_71725953843360
// MI455X (gfx1250) — compile-verified
//
#include <hip/hip_runtime.h>
#include <hip/hip_bf16.h>
#include <math.h>

typedef __attribute__((ext_vector_type(2))) float v2f;
typedef __attribute__((ext_vector_type(8))) float v8f;

#define NROWS 4096
#define FIN   512
#define FOUT  256

__device__ __forceinline__ float elu1(float x) {
    return x > 0.0f ? x : (__expf(x) - 1.0f);
}

// ---------------------------------------------------------------------------
// Kernel 1: h[m][n] = sum_k input_r[m][k] * W_r[n][k]   (h = input_r @ W_r.T)
// One wave per 16x16 tile of h, V_WMMA_F32_16X16X4_F32 over K=512 (128 WMMAs).
//
// A fragment (16x4 f32, 2 VGPR/lane): lanes 0-15 row M=lane hold K=0,1;
// lanes 16-31 row M=lane-16 hold K=2,3  -> float2 load at row*FIN + k + (lane>>4)*2.
// B fragment (4x16 f32): B[k][n] = W_r[n][k]; same per-lane addressing with
// W_r row = tile_n*16 + (lane&15).
// C/D (16x16 f32, 8 VGPR): reg r -> (M = r + (lane>=16)*8, N = lane&15).
// ---------------------------------------------------------------------------
__global__ __launch_bounds__(256) void k_gemm_h(const float* __restrict__ input_r,
                                                const float* __restrict__ W_r,
                                                float* __restrict__ h) {
    const int lane   = threadIdx.x & 31;
    const int wave   = threadIdx.x >> 5;
    const int tile_n = blockIdx.x;                 // 0..15   (FOUT/16)
    const int tile_m = blockIdx.y * 8 + wave;      // 0..255  (NROWS/16)

    const int half  = lane >> 4;                   // 0: K lo pair, 1: K hi pair
    const int mrow  = tile_m * 16 + (lane & 15);
    const int brow  = tile_n * 16 + (lane & 15);   // row of W_r (= output col)

    const float* Ap = input_r + (size_t)mrow * FIN + half * 2;
    const float* Bp = W_r     + (size_t)brow * FIN + half * 2;

    v8f acc = {};
#pragma unroll 4
    for (int k = 0; k < FIN; k += 4) {
        v2f a = *(const v2f*)(Ap + k);
        v2f b = *(const v2f*)(Bp + k);
        acc = __builtin_amdgcn_wmma_f32_16x16x4_f32(
            /*neg_a=*/false, a, /*neg_b=*/false, b,
            /*c_mod=*/(short)0, acc, /*reuse_a=*/false, /*reuse_b=*/false);
    }

    const int n_out  = tile_n * 16 + (lane & 15);
    const int m_base = tile_m * 16 + half * 8;
#pragma unroll
    for (int r = 0; r < 8; ++r)
        h[(size_t)(m_base + r) * FOUT + n_out] = acc[r];
}

// ---------------------------------------------------------------------------
// Kernel 2: f1[i] = h[i,:]@w_f1 ; f2[i] = h[i,:]@w_f2. One wave per row.
// ---------------------------------------------------------------------------
__global__ __launch_bounds__(256) void k_fvec(const float* __restrict__ h,
                                              const float* __restrict__ w_f1,
                                              const float* __restrict__ w_f2,
                                              float* __restrict__ f1,
                                              float* __restrict__ f2) {
    const int lane = threadIdx.x & 31;
    const int wave = threadIdx.x >> 5;
    const int row  = blockIdx.x * 8 + wave;

    float s1 = 0.0f, s2 = 0.0f;
    const float* hp = h + (size_t)row * FOUT;
#pragma unroll
    for (int n = lane; n < FOUT; n += 32) {
        float hv = hp[n];
        s1 = fmaf(hv, w_f1[n], s1);
        s2 = fmaf(hv, w_f2[n], s2);
    }
#pragma unroll
    for (int off = 16; off > 0; off >>= 1) {
        s1 += __shfl_xor(s1, off, 32);
        s2 += __shfl_xor(s2, off, 32);
    }
    if (lane == 0) { f1[row] = s1; f2[row] = s2; }
}

// ---------------------------------------------------------------------------
// Kernel 3: maxf1 = max(f1). Single block.
// ---------------------------------------------------------------------------
__global__ __launch_bounds__(256) void k_maxf1(const float* __restrict__ f1,
                                               float* __restrict__ maxf1) {
    __shared__ float red[256];
    const int tid = threadIdx.x;
    float m = -INFINITY;
    for (int i = tid; i < NROWS; i += 256) m = fmaxf(m, f1[i]);
    red[tid] = m;
    __syncthreads();
#pragma unroll
    for (int s = 128; s > 0; s >>= 1) {
        if (tid < s) red[tid] = fmaxf(red[tid], red[tid + s]);
        __syncthreads();
    }
    if (tid == 0) maxf1[0] = red[0];
}

// ---------------------------------------------------------------------------
// Kernel 4: one workgroup per row i.
//   s_j   = elu(f1[j] + f2[i]) / elu(maxf1 + f2[i])        (full-row normalizer)
//   p_j   = softmax over j>=i of s_j (j<i masked to -inf -> 0)
//   p_j   = p_j > 1/(N-i) ? p_j : 0 ;  p_i = 0 (diag)
//   out[i][j] = p_j * c[i][j]
// f1 staged in LDS; exp values cached in LDS to avoid a third transcendental
// pass. c / out streamed once -> non-temporal.
// ---------------------------------------------------------------------------
__global__ __launch_bounds__(256) void k_row_softmax(const float* __restrict__ f1,
                                                     const float* __restrict__ f2,
                                                     const float* __restrict__ maxf1,
                                                     const float* __restrict__ c,
                                                     float* __restrict__ out) {
    __shared__ float sF1[NROWS];
    __shared__ float sE[NROWS];
    __shared__ float red[256];

    const int tid = threadIdx.x;
    const int i   = blockIdx.x;

    for (int j = tid; j < NROWS; j += 256) sF1[j] = f1[j];
    const float f2i   = f2[i];
    const float inv_r = 1.0f / elu1(maxf1[0] + f2i);
    __syncthreads();

    // pass 1: masked row max of s_j, caching s_j in sE
    float m = -INFINITY;
    for (int j = i + tid; j < NROWS; j += 256) {
        float s = elu1(sF1[j] + f2i) * inv_r;
        sE[j] = s;
        m = fmaxf(m, s);
    }
    red[tid] = m;
    __syncthreads();
#pragma unroll
    for (int s = 128; s > 0; s >>= 1) {
        if (tid < s) red[tid] = fmaxf(red[tid], red[tid + s]);
        __syncthreads();
    }
    m = red[0];
    __syncthreads();

    // pass 2: exp + sum, exp cached back into sE
    float acc = 0.0f;
    for (int j = i + tid; j < NROWS; j += 256) {
        float e = __expf(sE[j] - m);
        sE[j] = e;
        acc += e;
    }
    red[tid] = acc;
    __syncthreads();
#pragma unroll
    for (int s = 128; s > 0; s >>= 1) {
        if (tid < s) red[tid] += red[tid + s];
        __syncthreads();
    }
    const float inv_sum = 1.0f / red[0];
    const float limit   = 1.0f / (float)(NROWS - i);

    // pass 3: normalize, threshold, zero diagonal, multiply by streamed c
    const size_t rowbase = (size_t)i * NROWS;
    for (int j = tid; j < NROWS; j += 256) {
        float p = 0.0f;
        if (j > i) {                       // j < i masked, j == i diag-zeroed
            p = sE[j] * inv_sum;
            p = (p > limit) ? p : 0.0f;
        }
        float cv = __builtin_nontemporal_load(c + rowbase + j);
        __builtin_nontemporal_store(p * cv, out + rowbase + j);
    }
}

extern "C" void kernel_launch(void* const* d_in, const int* in_sizes, int n_in,
                              void* d_out, int out_size, void* d_ws, size_t ws_size,
                              hipStream_t stream) {
    const float* input_r = (const float*)d_in[0];   // [N, F_IN]
    const float* c       = (const float*)d_in[1];   // [N, N]
    const float* W_r     = (const float*)d_in[2];   // [F_OUT, F_IN]
    const float* w_f1    = (const float*)d_in[3];   // [F_OUT]
    const float* w_f2    = (const float*)d_in[4];   // [F_OUT]
    float* out = (float*)d_out;                     // [N, N]

    float* h     = (float*)d_ws;                    // N*F_OUT floats (4 MB)
    float* f1    = h  + (size_t)NROWS * FOUT;       // N
    float* f2    = f1 + NROWS;                      // N
    float* maxf1 = f2 + NROWS;                      // 1

    k_gemm_h<<<dim3(FOUT / 16, NROWS / (16 * 8)), 256, 0, stream>>>(input_r, W_r, h);
    k_fvec<<<NROWS / 8, 256, 0, stream>>>(h, w_f1, w_f2, f1, f2);
    k_maxf1<<<1, 256, 0, stream>>>(f1, maxf1);
    k_row_softmax<<<NROWS, 256, 0, stream>>>(f1, f2, maxf1, c, out);
}